// MonoFlangerChorusModule_13993003450899
// MI455X (gfx1250) — compile-verified
//
#include <hip/hip_runtime.h>
#include <stdint.h>

// Flanger/chorus recurrence, one block (1 wave32) per batch, both channels.
//
// MIN_D=44, MAX_LFO=441, MAX_D=485 (44.1 kHz constants from the reference).
// delay >= 44 samples => any 32-sample window is internally dependence-free:
// all fractional-tap reads hit buffer state older than the window, and the
// window's writes land in 32 distinct circular-buffer slots. So each wave32
// iteration retires 32 time steps: all LDS reads, then all LDS writes
// (same-wave LDS ops are architecturally in-order; single-wave workgroup
// => no barriers at all).
//
// Global x / mod_sig traffic uses the CDNA5 async DMA path
// (global_load_async_to_lds_b32 / ASYNCcnt / s_wait_asynccnt) with an
// 8-slot ring and prefetch distance 7 (24 outstanding async ops max,
// ASYNCcnt is 6-bit), so HBM/L2 latency never sits on the serial DS chain.

#define F_MIN_D   44
#define F_MAX_LFO 441
#define F_MAX_D   485
#define F_CHUNK   32
#define F_NSTAGE  8           // staging ring slots (power of 2)
#define F_PF      7           // prefetch distance (chunks ahead)

__global__ __launch_bounds__(32)
void flanger_kernel(const float* __restrict__ x,
                    const float* __restrict__ mod_sig,
                    const float* __restrict__ feedback,
                    const float* __restrict__ width,
                    const float* __restrict__ depth,
                    const float* __restrict__ mix,
                    float* __restrict__ out,
                    int S)
{
    __shared__ float buf[2 * 512];                   // two delay lines, [0..484] used
    __shared__ float xs[F_NSTAGE][2][F_CHUNK];       // [stage][channel][lane]
    __shared__ float ms[F_NSTAGE][F_CHUNK];          // [stage][lane]

    const int b      = blockIdx.x;
    const int lane   = threadIdx.x;
    const int nchunk = S / F_CHUNK;

    // zero both delay lines (in-order with later same-wave LDS reads)
    for (int i = lane; i < 2 * 512; i += F_CHUNK)
        buf[i] = 0.0f;

    const float fb    = feedback[b];
    const float scale = (float)F_MAX_LFO * width[b];
    const float kiv   = mix[b] * depth[b];           // out = x + mix*depth*iv

    const float* xb0 = x       + (size_t)(b * 2 + 0) * (size_t)S;
    const float* xb1 = x       + (size_t)(b * 2 + 1) * (size_t)S;
    const float* mb  = mod_sig + (size_t)b * (size_t)S;
    float*       ob0 = out     + (size_t)(b * 2 + 0) * (size_t)S;
    float*       ob1 = out     + (size_t)(b * 2 + 1) * (size_t)S;

    // issue the 3 async global->LDS loads for one chunk into one ring slot
    auto issue = [&](int chunk) {
        const int s = chunk & (F_NSTAGE - 1);
        uint32_t voff = (uint32_t)((chunk * F_CHUNK + lane) * 4);
        uint32_t l0 = (uint32_t)(uintptr_t)&xs[s][0][lane];   // LDS byte offset
        uint32_t l1 = (uint32_t)(uintptr_t)&xs[s][1][lane];
        uint32_t lm = (uint32_t)(uintptr_t)&ms[s][lane];
        asm volatile("global_load_async_to_lds_b32 %0, %1, %2"
                     :: "v"(l0), "v"(voff), "s"(xb0) : "memory");
        asm volatile("global_load_async_to_lds_b32 %0, %1, %2"
                     :: "v"(l1), "v"(voff), "s"(xb1) : "memory");
        asm volatile("global_load_async_to_lds_b32 %0, %1, %2"
                     :: "v"(lm), "v"(voff), "s"(mb) : "memory");
    };

    int w = lane;                                    // write index = t % 485

    // one iteration of the recurrence window (32 time steps, both channels)
    auto body = [&](int chunk) {
        const int s = chunk & (F_NSTAGE - 1);
        const int t = chunk * F_CHUNK + lane;

        const float mv = ms[s][lane];
        const float x0 = xs[s][0][lane];
        const float x1 = xs[s][1][lane];

        // fractional read index into the circular buffer
        float delay = fmaf(scale, mv, (float)F_MIN_D);       // in [44, 485)
        float ridx  = (float)w - delay;
        if (ridx < 0.0f)            ridx += (float)F_MAX_D;
        if (ridx >= (float)F_MAX_D) ridx -= (float)F_MAX_D;
        float pf   = __builtin_floorf(ridx);
        float frac = ridx - pf;
        int pi = (int)pf;
        int ni = pi + 1; if (ni >= F_MAX_D) ni = 0;

        // --- all reads first (depend only on pre-window state: delay >= 44 > 31)
        float pv0 = buf[pi];
        float nv0 = buf[ni];
        float pv1 = buf[512 + pi];
        float nv1 = buf[512 + ni];

        float iv0 = fmaf(frac, nv0 - pv0, pv0);
        float iv1 = fmaf(frac, nv1 - pv1, pv1);

        // --- then all writes (distinct slots; in-order vs. the reads above)
        buf[w]       = fmaf(fb, iv0, x0);
        buf[512 + w] = fmaf(fb, iv1, x1);

        ob0[t] = fmaf(kiv, iv0, x0);
        ob1[t] = fmaf(kiv, iv1, x1);

        w += F_CHUNK; if (w >= F_MAX_D) w -= F_MAX_D;
    };

    // prologue: fill the prefetch pipeline (<= 21 outstanding async ops)
    for (int c = 0; c < F_PF && c < nchunk; ++c)
        issue(c);

    int chunk = 0;

    // steady state: issue chunk+7, retire oldest 3 async loads, run window.
    // After issue there are 24 outstanding; waiting to <=21 completes the
    // current chunk's 3 loads (async completion is in-order).
    for (; chunk + F_PF < nchunk; ++chunk) {
        issue(chunk + F_PF);
        asm volatile("s_wait_asynccnt 0x15" ::: "memory");   // 21 = 3*F_PF
        body(chunk);
    }

    // drain epilogue: last F_PF chunks, everything already issued
    for (; chunk < nchunk; ++chunk) {
        asm volatile("s_wait_asynccnt 0x0" ::: "memory");
        body(chunk);
    }
}

extern "C" void kernel_launch(void* const* d_in, const int* in_sizes, int n_in,
                              void* d_out, int out_size, void* d_ws, size_t ws_size,
                              hipStream_t stream) {
    const float* x        = (const float*)d_in[0];
    const float* mod_sig  = (const float*)d_in[1];
    const float* feedback = (const float*)d_in[2];
    const float* width    = (const float*)d_in[3];
    const float* depth    = (const float*)d_in[4];
    const float* mix      = (const float*)d_in[5];
    float* out            = (float*)d_out;

    const int B = in_sizes[2];              // feedback is (B,)
    const int S = in_sizes[1] / B;          // mod_sig is (B,S); S=131072 (mult of 32)

    flanger_kernel<<<B, 32, 0, stream>>>(x, mod_sig, feedback, width, depth, mix,
                                         out, S);
}